// RoiAlign_54563264529000
// MI455X (gfx1250) — compile-verified
//
#include <hip/hip_runtime.h>
#include <cstdint>
#include <cstddef>

// Native clang vector type (required by __builtin_nontemporal_store; also
// lowers cleanly to global_load_b128 / global_store_b128).
typedef float v4f __attribute__((ext_vector_type(4)));

// Problem constants (fixed by the reference's setup_inputs).
constexpr int kB  = 2;      // batch
constexpr int kN  = 2000;   // boxes per batch
constexpr int kNC = 4;      // classes
constexpr int kK  = 256;    // TOP_K
constexpr int kBK = kB * kK;   // 512 flat rois
constexpr int kCH = 14;     // crop h
constexpr int kCW = 14;     // crop w
constexpr int kC  = 256;    // channels

// ---------------------------------------------------------------------------
// 1) scores[b,n] = max_c classification[b,n,c]
// ---------------------------------------------------------------------------
__global__ void scores_kernel(const float* __restrict__ cls,
                              float* __restrict__ scores) {
  int i = blockIdx.x * blockDim.x + threadIdx.x;
  if (i >= kB * kN) return;
  const float* p = cls + (size_t)i * kNC;
  scores[i] = fmaxf(fmaxf(p[0], p[1]), fmaxf(p[2], p[3]));
}

// ---------------------------------------------------------------------------
// 2) top-K per batch via stable rank counting (matches jax.lax.top_k:
//    descending values, ties broken by smaller index).
// ---------------------------------------------------------------------------
__global__ void topk_kernel(const float* __restrict__ scores,
                            int* __restrict__ topk_idx) {
  int i = blockIdx.x * blockDim.x + threadIdx.x;
  if (i >= kB * kN) return;
  int b = i / kN, n = i - b * kN;
  const float* s = scores + (size_t)b * kN;
  float v = s[n];
  int rank = 0;
  for (int j = 0; j < kN; ++j) {
    float sj = s[j];
    rank += (int)((sj > v) || (sj == v && j < n));
  }
  if (rank < kK) topk_idx[b * kK + rank] = n;
}

// ---------------------------------------------------------------------------
// 3) Gather selected boxes, compute FPN level per roi.
//    lv = clip(floor(1 + log2(sqrt(w*h)/224 + 1e-7)), 0, 4)
// ---------------------------------------------------------------------------
__global__ void prep_kernel(const float* __restrict__ boxes,
                            const int* __restrict__ topk_idx,
                            float* __restrict__ selbox,
                            int* __restrict__ levels) {
  int r = blockIdx.x * blockDim.x + threadIdx.x;
  if (r >= kBK) return;
  int b = r / kK;
  int n = topk_idx[r];
  const float* bx = boxes + ((size_t)b * kN + n) * 4;
  float x1 = bx[0], y1 = bx[1], x2 = bx[2], y2 = bx[3];
  selbox[r * 4 + 0] = x1;
  selbox[r * 4 + 1] = y1;
  selbox[r * 4 + 2] = x2;
  selbox[r * 4 + 3] = y2;
  float sz = sqrtf((x2 - x1) * (y2 - y1));
  float lv = floorf(1.0f + log2f(sz / 224.0f + 1e-7f));
  lv = fminf(fmaxf(lv, 0.0f), 4.0f);
  levels[r] = (int)lv;
}

// ---------------------------------------------------------------------------
// 4) Stable sort position: pos[r] = #{j: lev[j]<lev[r]} + #{j<r: lev[j]==lev[r]}
//    (stable argsort of the flat level list; out[pos[r]] = in[r]).
// ---------------------------------------------------------------------------
__global__ void pos_kernel(const int* __restrict__ levels,
                           int* __restrict__ pos) {
  int r = blockIdx.x * blockDim.x + threadIdx.x;
  if (r >= kBK) return;
  int lr = levels[r];
  int p = 0;
  for (int j = 0; j < kBK; ++j) {
    int lj = levels[j];
    p += (int)((lj < lr) || (lj == lr && j < r));
  }
  pos[r] = p;
}

// ---------------------------------------------------------------------------
// 5) Scatter reordered boxes + classification into the output head.
//    d_out layout: [boxes 512*4][class 512*4][rois 512*14*14*256]
// ---------------------------------------------------------------------------
__global__ void small_out_kernel(const float* __restrict__ selbox,
                                 const int* __restrict__ topk_idx,
                                 const int* __restrict__ pos,
                                 const float* __restrict__ cls,
                                 float* __restrict__ out) {
  int r = blockIdx.x * blockDim.x + threadIdx.x;
  if (r >= kBK) return;
  int b = r / kK;
  int n = topk_idx[r];
  int p = pos[r];
  for (int c = 0; c < 4; ++c) out[(size_t)p * 4 + c] = selbox[r * 4 + c];
  const float* cp = cls + ((size_t)b * kN + n) * kNC;
  float* op = out + (size_t)kBK * 4 + (size_t)p * kNC;
  for (int c = 0; c < kNC; ++c) op[c] = cp[c];
}

// ---------------------------------------------------------------------------
// 6) Main ROI-Align kernel: one 256-thread block (8 wave32) per (roi, crop_y).
//    Thread t -> (x-phase = t>>6, channel base c = (t&63)*4):
//      * corner gathers are 128-bit coalesced (64 lanes x 16B = 1KB/row seg),
//        regular-temporal so they stay L2-resident (FPN maps ~178MB < 192MB L2)
//      * output is write-once -> streamed with non-temporal 128-bit stores
//        (global_store_b128 th:NT) so it does not evict the gather working set
//      * gfx1250 global_prefetch of the next crop-column's corner rows
//    Exact tf.image.crop_and_resize math (bilinear, extrapolation_value = 0).
// ---------------------------------------------------------------------------
__global__ void __launch_bounds__(256) roi_kernel(
    const float* __restrict__ image_shape,
    const float* __restrict__ selbox,
    const int* __restrict__ levels,
    const int* __restrict__ pos,
    const float* __restrict__ f0, const float* __restrict__ f1,
    const float* __restrict__ f2, const float* __restrict__ f3,
    const float* __restrict__ f4,
    float* __restrict__ out) {
  const int blk = blockIdx.x;
  const int r = blk / kCH;         // flat roi index
  const int y = blk - r * kCH;     // crop row
  const int b = r / kK;

  const int lv = levels[r];
  const float* feat;
  int S;
  switch (lv) {
    case 0:  feat = f0; S = 256; break;
    case 1:  feat = f1; S = 128; break;
    case 2:  feat = f2; S = 64;  break;
    case 3:  feat = f3; S = 32;  break;
    default: feat = f4; S = 16;  break;
  }
  const float fH = (float)S, fW = (float)S;
  const float img_h = image_shape[1];
  const float img_w = image_shape[2];

  const float bx1 = selbox[r * 4 + 0];
  const float by1 = selbox[r * 4 + 1];
  const float bx2 = selbox[r * 4 + 2];
  const float by2 = selbox[r * 4 + 3];

  // Exact normalization used by the module before crop_and_resize.
  const float y1n = by1 / img_h * fH / (fH - 1.0f);
  const float x1n = bx1 / img_w * fW / (fW - 1.0f);
  const float y2n = (by2 / img_h * fH - 1.0f) / (fH - 1.0f);
  const float x2n = (bx2 / img_w * fW - 1.0f) / (fW - 1.0f);

  const float ty  = (float)y / 13.0f;
  const float ysf = (y1n + ty * (y2n - y1n)) * (fH - 1.0f);
  const bool  vy  = (ysf >= 0.0f) && (ysf <= fH - 1.0f);
  const float y0f = floorf(ysf);
  const float fy  = ysf - y0f;
  const int y0i = (int)fminf(fmaxf(y0f, 0.0f), fH - 1.0f);
  const int y1i = (int)fminf(fmaxf(y0f + 1.0f, 0.0f), fH - 1.0f);

  const size_t head = (size_t)kBK * 4 + (size_t)kBK * kNC;  // 4096 floats
  float* orow = out + head + ((size_t)pos[r] * kCH + y) * (size_t)(kCW * kC);

  const int t = threadIdx.x;
  const int xp = t >> 6;          // 0..3  (x phase)
  const int c  = (t & 63) * 4;    // channel base (v4f)

  const float* row0 = feat + (size_t)(b * S + y0i) * S * kC;
  const float* row1 = feat + (size_t)(b * S + y1i) * S * kC;
  const float gy = 1.0f - fy;

  for (int it = 0; it < 4; ++it) {
    const int x = xp + it * 4;
    if (x >= kCW) break;  // x = 14,15 unused

    const float tx  = (float)x / 13.0f;
    const float xsf = (x1n + tx * (x2n - x1n)) * (fW - 1.0f);
    const bool  vx  = (xsf >= 0.0f) && (xsf <= fW - 1.0f);

    v4f o = (v4f)(0.0f);
    if (vy && vx) {
      const float x0f = floorf(xsf);
      const float fx  = xsf - x0f;
      const int x0i = (int)fminf(fmaxf(x0f, 0.0f), fW - 1.0f);
      const int x1i = (int)fminf(fmaxf(x0f + 1.0f, 0.0f), fW - 1.0f);

      // gfx1250 global_prefetch of next x's corner rows (L2-resident features).
      const int xn = x + 4;
      if (xn < kCW) {
        const float txn = (float)xn / 13.0f;
        const float xsn = (x1n + txn * (x2n - x1n)) * (fW - 1.0f);
        const int x0n = (int)fminf(fmaxf(floorf(xsn), 0.0f), fW - 1.0f);
        __builtin_prefetch(row0 + (size_t)x0n * kC + c, 0, 0);
        __builtin_prefetch(row1 + (size_t)x0n * kC + c, 0, 0);
      }

      const v4f v00 = *(const v4f*)(row0 + (size_t)x0i * kC + c);
      const v4f v01 = *(const v4f*)(row0 + (size_t)x1i * kC + c);
      const v4f v10 = *(const v4f*)(row1 + (size_t)x0i * kC + c);
      const v4f v11 = *(const v4f*)(row1 + (size_t)x1i * kC + c);

      const float gx = 1.0f - fx;
      o = (v00 * gx + v01 * fx) * gy + (v10 * gx + v11 * fx) * fy;
    }
    // Streaming (non-temporal) 128-bit store: output is never re-read, keep
    // L2 for the gather working set.
    __builtin_nontemporal_store(o, (v4f*)(orow + (size_t)x * kC + c));
  }
}

// ---------------------------------------------------------------------------
// Launch
// ---------------------------------------------------------------------------
extern "C" void kernel_launch(void* const* d_in, const int* in_sizes, int n_in,
                              void* d_out, int out_size, void* d_ws, size_t ws_size,
                              hipStream_t stream) {
  (void)in_sizes; (void)n_in; (void)out_size; (void)ws_size;

  const float* image_shape = (const float*)d_in[0];
  const float* boxes       = (const float*)d_in[1];
  const float* cls         = (const float*)d_in[2];
  const float* f0          = (const float*)d_in[3];
  const float* f1          = (const float*)d_in[4];
  const float* f2          = (const float*)d_in[5];
  const float* f3          = (const float*)d_in[6];
  const float* f4          = (const float*)d_in[7];
  float* out = (float*)d_out;

  // Workspace layout (all 16B-aligned, ~35 KB total).
  char* ws = (char*)d_ws;
  float* scores   = (float*)(ws);                      // 4000 f32
  int*   topk_idx = (int*)  (ws + 16384);              // 512 i32
  float* selbox   = (float*)(ws + 16384 + 4096);       // 2048 f32
  int*   levels   = (int*)  (ws + 16384 + 4096 + 8192);        // 512 i32
  int*   pos      = (int*)  (ws + 16384 + 4096 + 8192 + 4096); // 512 i32

  scores_kernel   <<<(kB * kN + 255) / 256, 256, 0, stream>>>(cls, scores);
  topk_kernel     <<<(kB * kN + 255) / 256, 256, 0, stream>>>(scores, topk_idx);
  prep_kernel     <<<(kBK + 255) / 256, 256, 0, stream>>>(boxes, topk_idx, selbox, levels);
  pos_kernel      <<<(kBK + 255) / 256, 256, 0, stream>>>(levels, pos);
  small_out_kernel<<<(kBK + 255) / 256, 256, 0, stream>>>(selbox, topk_idx, pos, cls, out);
  roi_kernel      <<<kBK * kCH, 256, 0, stream>>>(image_shape, selbox, levels, pos,
                                                  f0, f1, f2, f3, f4, out);
}